// BilateralOperator_56126632624322
// MI455X (gfx1250) — compile-verified
//
#include <hip/hip_runtime.h>

// Range-only bilateral filter, all-pairs per (batch,channel) plane.
// E[q,p] = c*(-xq^2 + 2*xq*xp - xp^2),  c = log2(e)/sigma^2  (base-2 exponent)
// Per 16x16 tile: one V_WMMA_F32_16X16X4_F32 builds the exponent tile,
// 8 v_exp_f32 per lane produce weights, fmac/add chains accumulate num/den.
// A-operand rows are precomputed into an LDS table so the inner loop carries
// no operand-construction VALU; high lane-half reads a stride-0 constant entry.

typedef __attribute__((ext_vector_type(2))) float v2f;
typedef __attribute__((ext_vector_type(8))) float v8f;

#define NPIX 4096                       // H*W = 64*64
#define CEXP 144.269504088896340f       // (1/sigma^2) * log2(e), sigma = 0.1

__device__ __forceinline__ float fast_exp2(float x) {
#if __has_builtin(__builtin_amdgcn_exp2f)
  return __builtin_amdgcn_exp2f(x);     // v_exp_f32
#else
  return exp2f(x);
#endif
}

__global__ __launch_bounds__(256) void bilateral_wmma_kernel(
    const float* __restrict__ x, float* __restrict__ out) {
  __shared__ float  xs[NPIX];           // 16 KB: channel plane (weights + B build)
  __shared__ float2 af[NPIX + 1];       // 32 KB: A rows (-c*xq^2, 2c*xq); [NPIX]=(-c,0)

  const int ch = blockIdx.y;
  const float* __restrict__ xc = x + ch * NPIX;

  // One-time stage: channel plane + precomputed A-operand table.
  {
    const float4* __restrict__ src = (const float4*)xc;
    float4* dstx = (float4*)xs;
    for (int i = threadIdx.x; i < NPIX / 4; i += (int)blockDim.x) {
      const float4 v = src[i];
      dstx[i] = v;
      const int j = i * 4;
      af[j + 0] = make_float2(-CEXP * v.x * v.x, 2.0f * CEXP * v.x);
      af[j + 1] = make_float2(-CEXP * v.y * v.y, 2.0f * CEXP * v.y);
      af[j + 2] = make_float2(-CEXP * v.z * v.z, 2.0f * CEXP * v.z);
      af[j + 3] = make_float2(-CEXP * v.w * v.w, 2.0f * CEXP * v.w);
    }
    if (threadIdx.x == 0) af[NPIX] = make_float2(-CEXP, 0.0f);  // hi-half K=2..3
  }
  __syncthreads();

  const int lane = threadIdx.x & 31;
  const int wave = threadIdx.x >> 5;
  const int col  = lane & 15;           // p within tile / q-row for A low half
  const int hi   = lane >> 4;           // lane half selects K-pairs / q-halves
  const int p0   = (blockIdx.x * (blockDim.x >> 5) + wave) * 16;

  // B operand (4x16, N=p): b_p = (1, xp, xp^2, 0)
  //   low lanes hold K=0..1 in VGPR0..1, high lanes hold K=2..3.
  const float xp = xs[p0 + col];
  v2f bmat;
  bmat[0] = hi ? xp * xp : 1.0f;        // K=2 : K=0
  bmat[1] = hi ? 0.0f    : xp;          // K=3 : K=1

  // Loop-carried pointers: low lanes walk the A table (stride 16 rows/tile),
  // high lanes pin on the constant entry (stride 0). Weight pointer walks xs.
  const float2* __restrict__ ap = af + (hi ? NPIX : col);
  const int astep = hi ? 0 : 16;
  const float4* __restrict__ wp = (const float4*)xs + (hi ? 2 : 0);

  float num0 = 0.0f, num1 = 0.0f, den0 = 0.0f, den1 = 0.0f;
  const v8f cz = {};

  for (int q0 = 0; q0 < NPIX; q0 += 16) {
    const float2 a = *ap;               // ds_load_b64, no VALU operand build
    ap += astep;
    v2f amat;
    amat[0] = a.x;
    amat[1] = a.y;

    // E^T tile in log2 domain: D[i] = E[q0 + i + 8*hi][p0 + col]
    v8f e = __builtin_amdgcn_wmma_f32_16x16x4_f32(
        false, amat, false, bmat, (short)0, cz, false, false);

    // 8 source-pixel weights for this lane's q-half (broadcast LDS reads).
    const float4 xa = wp[0];
    const float4 xb = wp[1];
    wp += 4;

    const float w0 = fast_exp2(e[0]);
    const float w1 = fast_exp2(e[1]);
    const float w2 = fast_exp2(e[2]);
    const float w3 = fast_exp2(e[3]);
    const float w4 = fast_exp2(e[4]);
    const float w5 = fast_exp2(e[5]);
    const float w6 = fast_exp2(e[6]);
    const float w7 = fast_exp2(e[7]);

    den0 += w0; den1 += w1; den0 += w2; den1 += w3;
    den0 += w4; den1 += w5; den0 += w6; den1 += w7;
    num0 = fmaf(w0, xa.x, num0); num1 = fmaf(w1, xa.y, num1);
    num0 = fmaf(w2, xa.z, num0); num1 = fmaf(w3, xa.w, num1);
    num0 = fmaf(w4, xb.x, num0); num1 = fmaf(w5, xb.y, num1);
    num0 = fmaf(w6, xb.z, num0); num1 = fmaf(w7, xb.w, num1);
  }

  float num = num0 + num1;
  float den = den0 + den1;

  // Merge the two q-halves: lane L <-> lane L^16 hold complementary sums.
  num += __shfl_xor(num, 16, 32);
  den += __shfl_xor(den, 16, 32);

  if (lane < 16) out[ch * NPIX + p0 + lane] = num / den;
}

extern "C" void kernel_launch(void* const* d_in, const int* in_sizes, int n_in,
                              void* d_out, int out_size, void* d_ws, size_t ws_size,
                              hipStream_t stream) {
  const float* x = (const float*)d_in[0];
  float* out = (float*)d_out;
  const int channels = in_sizes[0] / NPIX;       // B*C = 6
  dim3 grid(NPIX / (16 * 8), channels, 1);       // 8 wave32 p-tiles per block
  dim3 block(256, 1, 1);
  hipLaunchKernelGGL(bilateral_wmma_kernel, grid, block, 0, stream, x, out);
}